// MultiFourierDecoupling_2508260901448
// MI455X (gfx1250) — compile-verified
//
#include <hip/hip_runtime.h>
#include <cstdint>
#include <cstddef>

typedef float v2f __attribute__((ext_vector_type(2)));
typedef float v8f __attribute__((ext_vector_type(8)));

// ---- CDNA5 async global<->LDS copy path (guarded: falls back to scalar copies) ----
#if defined(__has_builtin)
#if __has_builtin(__builtin_amdgcn_global_load_async_to_lds_b32)
#define HAVE_ASYNC_LD 1
#endif
#if __has_builtin(__builtin_amdgcn_global_store_async_from_lds_b32)
#define HAVE_ASYNC_ST 1
#endif
#endif

// Builtin parameter types (from hipcc diagnostics): arg0 = addrspace(1) int* (global),
// arg1 = addrspace(3) int* (LDS), then two integer immediates (offset, cpol).
typedef __attribute__((address_space(1))) int gas_i32;
typedef __attribute__((address_space(3))) int las_i32;

#if defined(HAVE_ASYNC_LD) || defined(HAVE_ASYNC_ST)
__device__ __forceinline__ void wait_async0() {
#if __has_builtin(__builtin_amdgcn_s_wait_asynccnt)
  __builtin_amdgcn_s_wait_asynccnt(0);
#else
  asm volatile("s_wait_asynccnt 0" ::: "memory");
#endif
}
#endif

namespace {
constexpr int Dv = 64, Hv = 64, Wv = 64, Bv = 2, Cv = 32;
constexpr int Nv = 131072;
constexpr int VOL = Dv * Hv * Wv;          // 262144
constexpr int TOT = Bv * Cv * VOL;         // 16777216 floats (67 MB)
constexpr int ROWS = TOT / 64;             // 262144 rows per transform stage
constexpr float CUT = 0.25f;
constexpr float PI2 = 6.283185307179586f;
}

// Address of element (row r, position k along transform axis) for axis stride S = 1<<sh.
// Layout is (b, c, d, h, w) row-major; rows enumerate all indices except the transform axis.
__device__ __forceinline__ size_t elem_addr(int r, int k, int sh) {
  const int outer = r >> sh;
  const int inner = r & ((1 << sh) - 1);
  return ((size_t)outer << (sh + 6)) + ((size_t)k << sh) + (size_t)inner;
}

// Radial band gain at spectral index (d, h, w). band 0 = low, band 1 = mid.
__device__ __forceinline__ float band_gain(int d, int h, int w, int band) {
  float fx = (float)(d < 32 ? d : d - 64) * (1.0f / 64.0f);
  float fy = (float)(h < 32 ? h : h - 64) * (1.0f / 64.0f);
  float fz = (float)(w < 32 ? w : w - 64) * (1.0f / 64.0f);
  float kn = sqrtf(fx * fx + fy * fy + fz * fz);
  bool low = kn <= CUT;
  bool high = kn >= 1.0f - CUT;
  bool keep = (band == 0) ? low : (!low && !high);
  return keep ? 1.0f : 0.0f;
}

__global__ void zero_kernel(float* p, int n) {
  int i = blockIdx.x * 256 + threadIdx.x;
  if (i < n) p[i] = 0.0f;
}

// One thread per (point, channel): dense[b, c, z, y, x] = features[n, c]
__global__ void scatter_kernel(const float* __restrict__ feats,
                               const int* __restrict__ idx4,
                               float* __restrict__ dense) {
  int t = blockIdx.x * 256 + threadIdx.x;
  if (t >= Nv * Cv) return;
  int n = t >> 5, c = t & 31;
  int b = idx4[4 * n + 0], z = idx4[4 * n + 1], y = idx4[4 * n + 2], x = idx4[4 * n + 3];
  size_t addr = ((((size_t)b * Cv + c) * Dv + z) * Hv + y) * Wv + x;
  dense[addr] = feats[(size_t)n * Cv + c];
}

// ---------------- batched 64-point complex DFT along one axis, via f32 WMMA.
// Block = 256 threads (8 wave32). Block owns 32 rows; each wave computes a 16x16 complex tile.
// Twiddles exp(sign*2pi*i*kj/64)*scale generated in-register (__sincosf) into LDS.
// band >= 0: multiply the spectral band mask while staging the input tile (used on the
// first inverse stage, sh==12, where (k, inner) = (d, h*64+w)).
// WRITE_IM=false also skips the imaginary WMMA chain (final inverse stage only needs Re).
template <bool HAS_IM, bool WRITE_IM>
__global__ __launch_bounds__(256)
void dft_stage_kernel(const float* __restrict__ in_re, const float* __restrict__ in_im,
                      float* __restrict__ out_re, float* __restrict__ out_im,
                      int sh, float sign, float scale, int band) {
  __shared__ float sxre[32][65];
  __shared__ float sxim[32][65];
  __shared__ float smre[64][65];
  __shared__ float smim[64][65];

  const int tid = threadIdx.x;
  const int row0 = blockIdx.x * 32;

  // Generate twiddle matrix (64x64 complex) into LDS: exp(sign*2pi*i*k*j/64) * scale
  for (int idx = tid; idx < 64 * 64; idx += 256) {
    int k = idx >> 6, j = idx & 63;
    float ang = PI2 * (float)((k * j) & 63) * (1.0f / 64.0f);
    float s, c;
    __sincosf(ang, &s, &c);
    smre[k][j] = c * scale;
    smim[k][j] = sign * s * scale;
  }

  // Stage this block's 32x64 complex input tile into LDS (coalesced along memory order)
  for (int idx = tid; idx < 32 * 64; idx += 256) {
    int rl, k;
    if (sh == 0) { rl = idx >> 6; k = idx & 63; }
    else         { k = idx >> 5; rl = idx & 31; }
    size_t a = elem_addr(row0 + rl, k, sh);
    if (band >= 0) {
      // masked stage (sh==12): k = d, inner = h*64+w
      int inner = (row0 + rl) & 4095;
      float g = band_gain(k, inner >> 6, inner & 63, band);
      sxre[rl][k] = in_re[a] * g;
      if (HAS_IM) sxim[rl][k] = in_im[a] * g;
    } else {
#ifdef HAVE_ASYNC_LD
      __builtin_amdgcn_global_load_async_to_lds_b32(
          (gas_i32*)(in_re + a), (las_i32*)&sxre[rl][k], 0, 0);
      if (HAS_IM)
        __builtin_amdgcn_global_load_async_to_lds_b32(
            (gas_i32*)(in_im + a), (las_i32*)&sxim[rl][k], 0, 0);
#else
      sxre[rl][k] = in_re[a];
      if (HAS_IM) sxim[rl][k] = in_im[a];
#endif
    }
  }
#ifdef HAVE_ASYNC_LD
  if (band < 0) wait_async0();
#endif
  __syncthreads();

  const int wv = tid >> 5;        // wave 0..7
  const int lane = tid & 31;
  const int m = lane & 15;
  const int hi = lane >> 4;       // lanes 16..31 handle K+2 / M+8 halves
  const int rt = wv >> 2;         // row-tile 0..1
  const int jt = wv & 3;          // col-tile 0..3
  const int arow = rt * 16 + m;
  const int jcol = jt * 16 + m;

  v8f acc_re = {};
  v8f acc_im = {};
#pragma unroll
  for (int kc = 0; kc < 16; ++kc) {
    const int ka = kc * 4 + hi * 2;  // this lane's K pair within the chunk
    v2f a_re; a_re.x = sxre[arow][ka];     a_re.y = sxre[arow][ka + 1];
    v2f b_re; b_re.x = smre[ka][jcol];     b_re.y = smre[ka + 1][jcol];
    // (neg_a, A, neg_b, B, c_mod, C, reuse_a, reuse_b)
    acc_re = __builtin_amdgcn_wmma_f32_16x16x4_f32(false, a_re, false, b_re, (short)0, acc_re, false, false);
    if (HAS_IM || WRITE_IM) {
      v2f b_im; b_im.x = smim[ka][jcol];   b_im.y = smim[ka + 1][jcol];
      if (WRITE_IM)
        acc_im = __builtin_amdgcn_wmma_f32_16x16x4_f32(false, a_re, false, b_im, (short)0, acc_im, false, false);
      if (HAS_IM) {
        v2f a_im;  a_im.x  = sxim[arow][ka];  a_im.y  = sxim[arow][ka + 1];
        v2f a_imn; a_imn.x = -a_im.x;         a_imn.y = -a_im.y;
        acc_re = __builtin_amdgcn_wmma_f32_16x16x4_f32(false, a_imn, false, b_im, (short)0, acc_re, false, false);
        if (WRITE_IM)
          acc_im = __builtin_amdgcn_wmma_f32_16x16x4_f32(false, a_im, false, b_re, (short)0, acc_im, false, false);
      }
    }
  }

  // Restage results through LDS for coalesced global stores
  __syncthreads();
#pragma unroll
  for (int v = 0; v < 8; ++v) {
    int rl = rt * 16 + hi * 8 + v;   // C/D layout: VGPR v -> M = v (+8 for upper lanes)
    sxre[rl][jcol] = acc_re[v];
    if (WRITE_IM) sxim[rl][jcol] = acc_im[v];
  }
  __syncthreads();
  for (int idx = tid; idx < 32 * 64; idx += 256) {
    int rl, k;
    if (sh == 0) { rl = idx >> 6; k = idx & 63; }
    else         { k = idx >> 5; rl = idx & 31; }
    size_t a = elem_addr(row0 + rl, k, sh);
#ifdef HAVE_ASYNC_ST
    __builtin_amdgcn_global_store_async_from_lds_b32(
        (gas_i32*)(out_re + a), (las_i32*)&sxre[rl][k], 0, 0);
    if (WRITE_IM)
      __builtin_amdgcn_global_store_async_from_lds_b32(
          (gas_i32*)(out_im + a), (las_i32*)&sxim[rl][k], 0, 0);
#else
    out_re[a] = sxre[rl][k];
    if (WRITE_IM) out_im[a] = sxim[rl][k];
#endif
  }
#ifdef HAVE_ASYNC_ST
  wait_async0();
#endif
}

__global__ void gather_kernel(const float* __restrict__ vol_re,
                              const int* __restrict__ idx4,
                              float* __restrict__ out) {
  int t = blockIdx.x * 256 + threadIdx.x;
  if (t >= Nv * Cv) return;
  int n = t >> 5, c = t & 31;
  int b = idx4[4 * n + 0], z = idx4[4 * n + 1], y = idx4[4 * n + 2], x = idx4[4 * n + 3];
  size_t addr = ((((size_t)b * Cv + c) * Dv + z) * Hv + y) * Wv + x;
  out[(size_t)n * Cv + c] = vol_re[addr];
}

// high = features - low - mid  (bands partition the spectrum -> skip third inverse FFT chain)
__global__ void high_kernel(const float* __restrict__ feats, float* __restrict__ out) {
  int t = blockIdx.x * 256 + threadIdx.x;
  if (t >= Nv * Cv) return;
  const size_t NC = (size_t)Nv * Cv;
  out[2 * NC + t] = feats[t] - out[t] - out[NC + t];
}

extern "C" void kernel_launch(void* const* d_in, const int* in_sizes, int n_in,
                              void* d_out, int out_size, void* d_ws, size_t ws_size,
                              hipStream_t stream) {
  const float* feats = (const float*)d_in[0];
  const int* idx4 = (const int*)d_in[1];
  float* out = (float*)d_out;
  float* ws = (float*)d_ws;

  // Workspace (floats): A = spectrum (in-place forward FFT), B = per-band inverse buffer
  float* A_re = ws;
  float* A_im = ws + (size_t)TOT;
  float* B_re = ws + 2 * (size_t)TOT;
  float* B_im = ws + 3 * (size_t)TOT;

  dim3 blk(256);
  zero_kernel<<<TOT / 256, blk, 0, stream>>>(A_re, TOT);
  scatter_kernel<<<(Nv * Cv) / 256, blk, 0, stream>>>(feats, idx4, A_re);

  // Forward 3D FFT, in place on A: W (stride 1), H (stride 64), D (stride 4096)
  dft_stage_kernel<false, true><<<ROWS / 32, blk, 0, stream>>>(A_re, A_im, A_re, A_im, 0, -1.0f, 1.0f, -1);
  dft_stage_kernel<true,  true><<<ROWS / 32, blk, 0, stream>>>(A_re, A_im, A_re, A_im, 6, -1.0f, 1.0f, -1);
  dft_stage_kernel<true,  true><<<ROWS / 32, blk, 0, stream>>>(A_re, A_im, A_re, A_im, 12, -1.0f, 1.0f, -1);

  // Low and mid bands: masked inv-D (A->B, mask fused into tile staging), then inv-H,
  // then inv-W (real output only), then gather. high = feats - low - mid.
  for (int band = 0; band < 2; ++band) {
    dft_stage_kernel<true, true ><<<ROWS / 32, blk, 0, stream>>>(A_re, A_im, B_re, B_im, 12, 1.0f, 1.0f / 64.0f, band);
    dft_stage_kernel<true, true ><<<ROWS / 32, blk, 0, stream>>>(B_re, B_im, B_re, B_im, 6,  1.0f, 1.0f / 64.0f, -1);
    dft_stage_kernel<true, false><<<ROWS / 32, blk, 0, stream>>>(B_re, B_im, B_re, B_im, 0,  1.0f, 1.0f / 64.0f, -1);
    gather_kernel<<<(Nv * Cv) / 256, blk, 0, stream>>>(B_re, idx4, out + (size_t)band * Nv * Cv);
  }
  high_kernel<<<(Nv * Cv) / 256, blk, 0, stream>>>(feats, out);
}